// SoftSilhouetteShader_16406775070840
// MI455X (gfx1250) — compile-verified
//
#include <hip/hip_runtime.h>
#include <hip/hip_bf16.h>
#include <stdint.h>

// Soft silhouette blend:  alpha = 1 - prod_k (1 - sigmoid(-d_k/sigma)*mask_k)
//                               = 1 - prod_k (mask_k ? sigmoid(d_k/sigma) : 1)
// sigma = 1e-4.  Output per pixel: {1,1,1,alpha}.
//
// Memory-bound: ~437 MB total traffic -> ~19 us floor at 23.3 TB/s.
// Strategy: stage each 128-pixel tile (dists + pix_to_face) into LDS with
// fully-coalesced gfx1250 async b128 copies (global_load_async_to_lds_b128),
// then reduce per-pixel from LDS with conflict-free ds_load_b64 reads.

#define THREADS     128
#define PIX         128                     // pixels per block
#define KFACE       50
#define ROW_BYTES   (KFACE * 4)             // 200 B per pixel per array
#define TILE_BYTES  (PIX * ROW_BYTES)       // 25600 B per array
#define CHUNKS      (TILE_BYTES / 16)       // 1600 b128 chunks per array

// CDNA5 async copy: global -> LDS, 16B per enabled lane, tracked by ASYNCcnt.
__device__ __forceinline__ void async_load_b128(uint32_t lds_byte_addr,
                                                const void* gaddr) {
    asm volatile("global_load_async_to_lds_b128 %0, %1, off"
                 :
                 : "v"(lds_byte_addr), "v"((uint64_t)(uintptr_t)gaddr)
                 : "memory");
}

__device__ __forceinline__ void wait_asynccnt0() {
    asm volatile("s_wait_asynccnt 0x0" ::: "memory");
}

// term = mask ? sigmoid(d/sigma) : 1
//      = mask ? rcp(1 + exp2(d * (-log2(e)/sigma))) : 1
// Saturates to exactly 0 / 1 for |d/sigma| large (inf -> rcp -> 0), matching
// the reference's saturated sigmoid within fp32 tolerance.
__device__ __forceinline__ float silh_term(float d, int face) {
    constexpr float C = -14426.950408889634f;     // -(1/sigma) * log2(e)
    float e = __builtin_amdgcn_exp2f(d * C);      // exp(-d/sigma)
    float s = __builtin_amdgcn_rcpf(1.0f + e);    // sigmoid(d/sigma)
    return (face >= 0) ? s : 1.0f;
}

__global__ void __launch_bounds__(THREADS, 1)
soft_silhouette_kernel(const float* __restrict__ dists,
                       const int* __restrict__ faces,
                       float* __restrict__ out,
                       int pixels) {
    __shared__ __align__(16) float ds_d[PIX * KFACE];   // 25.6 KB
    __shared__ __align__(16) int   ds_f[PIX * KFACE];   // 25.6 KB

    const int tid  = threadIdx.x;
    const int base = blockIdx.x * PIX;
    const int valid = pixels - base;            // uniform per block

    if (valid >= PIX) {
        // ---- Stage tile into LDS with coalesced async b128 copies ----
        const char* gd = (const char*)dists + (size_t)base * ROW_BYTES;
        const char* gf = (const char*)faces + (size_t)base * ROW_BYTES;
        // Low 32 bits of a generic pointer to LDS == workgroup-relative
        // LDS byte offset (aperture bits live in [63:32]).
        const uint32_t ld = (uint32_t)(uintptr_t)&ds_d[0];
        const uint32_t lf = (uint32_t)(uintptr_t)&ds_f[0];

        for (int i = tid; i < CHUNKS; i += THREADS) {
            async_load_b128(ld + (uint32_t)i * 16u, gd + (size_t)i * 16);
            async_load_b128(lf + (uint32_t)i * 16u, gf + (size_t)i * 16);
        }
        wait_asynccnt0();
        __syncthreads();

        // ---- Per-pixel product reduction from LDS ----
        // Lane stride = 50 dwords; gcd(50,64)=2 => 32 lanes hit 32 distinct
        // even banks; each b64 also uses bank+1 => all 64 banks used once:
        // conflict-free ds_load_b64.
        const float2* dl = reinterpret_cast<const float2*>(&ds_d[tid * KFACE]);
        const int2*   fl = reinterpret_cast<const int2*>(&ds_f[tid * KFACE]);

        float prod = 1.0f;
#pragma unroll
        for (int k = 0; k < KFACE / 2; ++k) {
            float2 d2 = dl[k];
            int2   f2 = fl[k];
            prod *= silh_term(d2.x, f2.x);
            prod *= silh_term(d2.y, f2.y);
        }

        const float alpha = 1.0f - prod;
        reinterpret_cast<float4*>(out)[base + tid] =
            make_float4(1.0f, 1.0f, 1.0f, alpha);
    } else {
        // Tail block (not hit for 4x512x512, but keep it correct/general):
        // direct global loads, no LDS staging.
        const int p = base + tid;
        if (p < pixels) {
            const float* dp = dists + (size_t)p * KFACE;
            const int*   fp = faces + (size_t)p * KFACE;
            float prod = 1.0f;
#pragma unroll
            for (int k = 0; k < KFACE; ++k)
                prod *= silh_term(dp[k], fp[k]);
            reinterpret_cast<float4*>(out)[p] =
                make_float4(1.0f, 1.0f, 1.0f, 1.0f - prod);
        }
    }
}

extern "C" void kernel_launch(void* const* d_in, const int* in_sizes, int n_in,
                              void* d_out, int out_size, void* d_ws, size_t ws_size,
                              hipStream_t stream) {
    (void)in_sizes; (void)n_in; (void)d_ws; (void)ws_size;
    const float* dists = (const float*)d_in[0];
    const int*   faces = (const int*)d_in[1];
    float*       out   = (float*)d_out;

    const int pixels = out_size / 4;                 // N*H*W
    const int blocks = (pixels + PIX - 1) / PIX;     // 8192 for 4x512x512

    soft_silhouette_kernel<<<blocks, THREADS, 0, stream>>>(dists, faces, out,
                                                           pixels);
}